// DiscourseBert_46471546143271
// MI455X (gfx1250) — compile-verified
//
#include <hip/hip_runtime.h>
#include <math.h>

#define B_ 8
#define L_ 128
#define H_ 768
#define P_ 16
#define DCOL 102
#define EPSF 1e-8f
#define NINF -1e30f
#define LDSTR 20          // LDS row stride (floats): conflict-free & 16B aligned
#define NCHUNK (H_ / 16)  // 48 K-chunks of 16

typedef float v2f __attribute__((ext_vector_type(2)));
typedef float v8f __attribute__((ext_vector_type(8)));

__device__ __forceinline__ v8f wmma4(v2f a, v2f b, v8f c) {
  // D = A(16x4 f32) x B(4x16 f32) + C(16x16 f32)
  return __builtin_amdgcn_wmma_f32_16x16x4_f32(false, a, false, b, (short)0, c,
                                               false, false);
}

// ---------------------------------------------------------------- prep: c = ctx*mask, row norms
__global__ void prep_kernel(const float* __restrict__ ctx1, const float* __restrict__ m1,
                            const float* __restrict__ ctx2, const float* __restrict__ m2,
                            float* c1, float* c2, float* n1, float* n2) {
  int row = blockIdx.x;            // b*L + i
  int side = blockIdx.y;
  const float* ctx = side ? ctx2 : ctx1;
  const float* m   = side ? m2 : m1;
  float* c = side ? c2 : c1;
  float* n = side ? n2 : n1;
  float msk = m[row];
  float ss = 0.f;
  for (int h = threadIdx.x; h < H_; h += 256) {
    float v = ctx[row * H_ + h] * msk;
    c[row * H_ + h] = v;
    ss += v * v;
  }
  __shared__ float red[256];
  red[threadIdx.x] = ss;
  __syncthreads();
  for (int s = 128; s > 0; s >>= 1) {
    if (threadIdx.x < s) red[threadIdx.x] += red[threadIdx.x + s];
    __syncthreads();
  }
  if (threadIdx.x == 0) n[row] = sqrtf(red[0]);
}

// lens[0..7]=len1, [8..15]=len2, [16..23]=lp1, [24..31]=lp2
__global__ void lens_kernel(const float* __restrict__ m1, const float* __restrict__ m2,
                            float* lens) {
  int t = threadIdx.x;
  if (t < 16) {
    int side = t >> 3, b = t & 7;
    const float* m = side ? m2 : m1;
    float s = 0.f;
    for (int j = 0; j < L_; ++j) s += m[b * L_ + j];
    lens[side * 8 + b] = s;
    int lp = (int)(s + 0.5f) - 1;
    if (lp < 0) lp = 0;
    lens[16 + side * 8 + b] = (float)lp;
  }
}

__global__ void w2_kernel(const float* __restrict__ w, float* w2) {
  int idx = blockIdx.x * 256 + threadIdx.x;
  if (idx < P_ * H_) { float v = w[idx]; w2[idx] = v * v; }
}

// ---------------------------------------------------------------- weighted row norms wn[b,p,i]
__global__ void wn_kernel(const float* __restrict__ c1, const float* __restrict__ c2,
                          const float* __restrict__ w2mp, float* wn1, float* wn2) {
  int row = blockIdx.x;            // b*L + i
  int side = blockIdx.y;
  const float* c = side ? c2 : c1;
  float* wn = side ? wn2 : wn1;
  __shared__ float sq[H_];
  for (int h = threadIdx.x; h < H_; h += 256) {
    float v = c[row * H_ + h];
    sq[h] = v * v;
  }
  __syncthreads();
  int p = threadIdx.x >> 4, l = threadIdx.x & 15;
  float s = 0.f;
  for (int h = l; h < H_; h += 16) s += w2mp[p * H_ + h] * sq[h];
  for (int off = 8; off >= 1; off >>= 1) s += __shfl_xor(s, off, 16);
  if (l == 0) {
    int b = row >> 7, i = row & 127;
    wn[(b * P_ + p) * L_ + i] = sqrtf(s);
  }
}

// ---------------------------------------------------------------- cos_sim GEMM (WMMA f32, LDS dbuf)
__global__ void cos_gemm_kernel(const float* __restrict__ c1, const float* __restrict__ c2,
                                const float* __restrict__ n1, const float* __restrict__ n2,
                                float* cosm) {
  int b = blockIdx.x;
  __shared__ float sa[2][L_ * LDSTR];
  __shared__ float sb[2][L_ * LDSTR];
  int t = threadIdx.x;
  int lane = t & 31, wv = t >> 5;
  int half = lane >> 4, l16 = lane & 15;
  int klo = half * 2;
  int i0 = wv * 16;
  int lrow = t >> 1, lko = (t & 1) * 8;   // loader mapping: 8 floats each of A & B
  const float* Ag = c1 + (size_t)(b * L_ + lrow) * H_ + lko;
  const float* Bg = c2 + (size_t)(b * L_ + lrow) * H_ + lko;

  float4 ra0 = *(const float4*)(Ag);
  float4 ra1 = *(const float4*)(Ag + 4);
  float4 rb0 = *(const float4*)(Bg);
  float4 rb1 = *(const float4*)(Bg + 4);
  *(float4*)&sa[0][lrow * LDSTR + lko]     = ra0;
  *(float4*)&sa[0][lrow * LDSTR + lko + 4] = ra1;
  *(float4*)&sb[0][lrow * LDSTR + lko]     = rb0;
  *(float4*)&sb[0][lrow * LDSTR + lko + 4] = rb1;
  __syncthreads();

  v8f acc[8] = {};
  for (int c = 0; c < NCHUNK; ++c) {
    int cur = c & 1, nxt = cur ^ 1;
    if (c + 1 < NCHUNK) {   // register prefetch of next chunk (latency hidden by WMMAs)
      int kc = (c + 1) * 16;
      ra0 = *(const float4*)(Ag + kc);
      ra1 = *(const float4*)(Ag + kc + 4);
      rb0 = *(const float4*)(Bg + kc);
      rb1 = *(const float4*)(Bg + kc + 4);
    }
    const float* pa = &sa[cur][(i0 + l16) * LDSTR + klo];
#pragma unroll
    for (int kk = 0; kk < 16; kk += 4) {
      float2 af = *(const float2*)(pa + kk);
      v2f a; a[0] = af.x; a[1] = af.y;
#pragma unroll
      for (int jt = 0; jt < 8; ++jt) {
        float2 bf = *(const float2*)(&sb[cur][(jt * 16 + l16) * LDSTR + kk + klo]);
        v2f bb; bb[0] = bf.x; bb[1] = bf.y;
        acc[jt] = wmma4(a, bb, acc[jt]);
      }
    }
    __syncthreads();
    if (c + 1 < NCHUNK) {
      *(float4*)&sa[nxt][lrow * LDSTR + lko]     = ra0;
      *(float4*)&sa[nxt][lrow * LDSTR + lko + 4] = ra1;
      *(float4*)&sb[nxt][lrow * LDSTR + lko]     = rb0;
      *(float4*)&sb[nxt][lrow * LDSTR + lko + 4] = rb1;
    }
    __syncthreads();
  }

  float rn1[8];
#pragma unroll
  for (int v = 0; v < 8; ++v) rn1[v] = fmaxf(n1[b * L_ + i0 + v + half * 8], EPSF);
#pragma unroll
  for (int jt = 0; jt < 8; ++jt) {
    int j = jt * 16 + l16;
    float rn2 = fmaxf(n2[b * L_ + j], EPSF);
#pragma unroll
    for (int v = 0; v < 8; ++v) {
      int i = i0 + v + half * 8;
      cosm[(size_t)(b * L_ + i) * L_ + j] = acc[jt][v] / (rn1[v] * rn2);
    }
  }
}

// ---------------------------------------------------------------- cos masked max/mean -> cols 0,1
__global__ void cosred_kernel(const float* __restrict__ cosm, const float* __restrict__ m1,
                              const float* __restrict__ m2, const float* __restrict__ lens,
                              float* out) {
  int r = blockIdx.x, b = blockIdx.y, side = blockIdx.z;
  int t = threadIdx.x;  // 128 threads
  float v, mk, lenf;
  if (side == 0) { v = cosm[(size_t)(b * L_ + r) * L_ + t]; mk = m2[b * L_ + t]; lenf = lens[8 + b]; }
  else           { v = cosm[(size_t)(b * L_ + t) * L_ + r]; mk = m1[b * L_ + t]; lenf = lens[b]; }
  __shared__ float rm[128], rs[128];
  rm[t] = (mk != 0.f) ? v : NINF;
  rs[t] = (mk != 0.f) ? v : 0.f;
  __syncthreads();
  for (int s = 64; s > 0; s >>= 1) {
    if (t < s) { rm[t] = fmaxf(rm[t], rm[t + s]); rs[t] += rs[t + s]; }
    __syncthreads();
  }
  if (t == 0) {
    int row = (side == 0) ? r : 128 + r;
    float* o = out + ((size_t)b * 256 + row) * DCOL;
    o[0] = rm[0];
    o[1] = rs[0] / fmaxf(lenf, EPSF);
  }
}

// ---------------------------------------------------------------- weighted pairwise GEMM (dominant)
__global__ void mp_gemm_kernel(const float* __restrict__ c1, const float* __restrict__ c2,
                               const float* __restrict__ w2mp,
                               const float* __restrict__ wn1, const float* __restrict__ wn2,
                               const float* __restrict__ m1, const float* __restrict__ m2,
                               const float* __restrict__ lens,
                               float* out, float* colmax, float* colsum) {
  int bp = blockIdx.x;             // b*16 + p
  int b = bp >> 4, p = bp & 15;
  __shared__ float sa[2][L_ * LDSTR];
  __shared__ float sb[2][L_ * LDSTR];
  int t = threadIdx.x;
  int lane = t & 31, wv = t >> 5;
  int half = lane >> 4, l16 = lane & 15;
  int klo = half * 2;
  int i0 = wv * 16;
  int lrow = t >> 1, lko = (t & 1) * 8;
  const float* Ag = c1 + (size_t)(b * L_ + lrow) * H_ + lko;
  const float* Bg = c2 + (size_t)(b * L_ + lrow) * H_ + lko;
  const float* Wg = w2mp + p * H_ + lko;

  float4 ra0 = *(const float4*)(Ag);
  float4 ra1 = *(const float4*)(Ag + 4);
  float4 rb0 = *(const float4*)(Bg);
  float4 rb1 = *(const float4*)(Bg + 4);
  float4 rw0 = *(const float4*)(Wg);
  float4 rw1 = *(const float4*)(Wg + 4);
  rb0.x *= rw0.x; rb0.y *= rw0.y; rb0.z *= rw0.z; rb0.w *= rw0.w;
  rb1.x *= rw1.x; rb1.y *= rw1.y; rb1.z *= rw1.z; rb1.w *= rw1.w;
  *(float4*)&sa[0][lrow * LDSTR + lko]     = ra0;
  *(float4*)&sa[0][lrow * LDSTR + lko + 4] = ra1;
  *(float4*)&sb[0][lrow * LDSTR + lko]     = rb0;
  *(float4*)&sb[0][lrow * LDSTR + lko + 4] = rb1;
  __syncthreads();

  v8f acc[8] = {};
  for (int c = 0; c < NCHUNK; ++c) {
    int cur = c & 1, nxt = cur ^ 1;
    if (c + 1 < NCHUNK) {
      int kc = (c + 1) * 16;
      ra0 = *(const float4*)(Ag + kc);
      ra1 = *(const float4*)(Ag + kc + 4);
      rb0 = *(const float4*)(Bg + kc);
      rb1 = *(const float4*)(Bg + kc + 4);
      rw0 = *(const float4*)(Wg + kc);
      rw1 = *(const float4*)(Wg + kc + 4);
    }
    const float* pa = &sa[cur][(i0 + l16) * LDSTR + klo];
#pragma unroll
    for (int kk = 0; kk < 16; kk += 4) {
      float2 af = *(const float2*)(pa + kk);
      v2f a; a[0] = af.x; a[1] = af.y;
#pragma unroll
      for (int jt = 0; jt < 8; ++jt) {
        float2 bf = *(const float2*)(&sb[cur][(jt * 16 + l16) * LDSTR + kk + klo]);
        v2f bb; bb[0] = bf.x; bb[1] = bf.y;
        acc[jt] = wmma4(a, bb, acc[jt]);
      }
    }
    __syncthreads();
    if (c + 1 < NCHUNK) {
      rb0.x *= rw0.x; rb0.y *= rw0.y; rb0.z *= rw0.z; rb0.w *= rw0.w;
      rb1.x *= rw1.x; rb1.y *= rw1.y; rb1.z *= rw1.z; rb1.w *= rw1.w;
      *(float4*)&sa[nxt][lrow * LDSTR + lko]     = ra0;
      *(float4*)&sa[nxt][lrow * LDSTR + lko + 4] = ra1;
      *(float4*)&sb[nxt][lrow * LDSTR + lko]     = rb0;
      *(float4*)&sb[nxt][lrow * LDSTR + lko + 4] = rb1;
    }
    __syncthreads();
  }

  float rn1[8], mk1[8];
#pragma unroll
  for (int v = 0; v < 8; ++v) {
    int i = i0 + v + half * 8;
    rn1[v] = wn1[(b * P_ + p) * L_ + i];
    mk1[v] = m1[b * L_ + i];
  }
  float rn2[8], mk2[8];
#pragma unroll
  for (int jt = 0; jt < 8; ++jt) {
    int j = jt * 16 + l16;
    rn2[jt] = wn2[(b * P_ + p) * L_ + j];
    mk2[jt] = m2[b * L_ + j];
  }
#pragma unroll
  for (int jt = 0; jt < 8; ++jt)
#pragma unroll
    for (int v = 0; v < 8; ++v)
      acc[jt][v] = acc[jt][v] / fmaxf(rn1[v] * rn2[jt], EPSF);

  float len2f = lens[8 + b];
  // row reduction over j (mask2) -> mv1_max / mv1_mean
#pragma unroll
  for (int v = 0; v < 8; ++v) {
    float m = NINF, s = 0.f;
#pragma unroll
    for (int jt = 0; jt < 8; ++jt) {
      float x = acc[jt][v];
      if (mk2[jt] != 0.f) { m = fmaxf(m, x); s += x; }
    }
    for (int off = 8; off >= 1; off >>= 1) {
      m = fmaxf(m, __shfl_xor(m, off, 16));
      s += __shfl_xor(s, off, 16);
    }
    if (l16 == 0) {
      int i = i0 + v + half * 8;
      float* o = out + ((size_t)b * 256 + i) * DCOL;
      o[36 + p] = m;
      o[52 + p] = s / fmaxf(len2f, EPSF);
    }
  }
  // column partials over this 16-row strip (mask1) -> mv2 via finish kernel
#pragma unroll
  for (int jt = 0; jt < 8; ++jt) {
    float m = NINF, s = 0.f;
#pragma unroll
    for (int v = 0; v < 8; ++v) {
      float x = acc[jt][v];
      if (mk1[v] != 0.f) { m = fmaxf(m, x); s += x; }
    }
    m = fmaxf(m, __shfl_xor(m, 16, 32));
    s += __shfl_xor(s, 16, 32);
    if (half == 0) {
      int j = jt * 16 + l16;
      int idx = (bp * 8 + wv) * L_ + j;
      colmax[idx] = m;
      colsum[idx] = s;
    }
  }
}

__global__ void colfinish_kernel(const float* __restrict__ colmax,
                                 const float* __restrict__ colsum,
                                 const float* __restrict__ lens, float* out) {
  int idx = blockIdx.x * 256 + threadIdx.x;   // (b*16+p)*128 + j
  if (idx >= B_ * P_ * L_) return;
  int j = idx & 127;
  int p = (idx >> 7) & 15;
  int b = idx >> 11;
  float m = NINF, s = 0.f;
  for (int w = 0; w < 8; ++w) {
    int q = ((b * P_ + p) * 8 + w) * L_ + j;
    m = fmaxf(m, colmax[q]);
    s += colsum[q];
  }
  float* o = out + ((size_t)b * 256 + 128 + j) * DCOL;
  o[36 + p] = m;
  o[52 + p] = s / fmaxf(lens[b], EPSF);
}

// ---------------------------------------------------------------- attention sum GEMMs (WMMA f32)
__global__ void attsum_gemm_kernel(const float* __restrict__ cosm,
                                   const float* __restrict__ c1, const float* __restrict__ c2,
                                   float* asum2, float* asum1) {
  int n0 = blockIdx.x * 128;
  int b = blockIdx.y;
  int side = blockIdx.z;
  int lane = threadIdx.x & 31, wv = threadIdx.x >> 5;
  int half = lane >> 4, l16 = lane & 15;
  int klo = half * 2;
  int r0 = wv * 16;
  const float* Bsrc = (side == 0) ? c2 : c1;
  float* dst = (side == 0) ? asum2 : asum1;
  v8f acc[8] = {};
  for (int kc = 0; kc < L_; kc += 4) {
    v2f a;
    if (side == 0) {  // A[i,k=j] = cos[b,i,j]  (contiguous in k)
      float2 af = *reinterpret_cast<const float2*>(
          cosm + (size_t)(b * L_ + r0 + l16) * L_ + kc + klo);
      a[0] = af.x; a[1] = af.y;
    } else {          // A[j,k=i] = cos[b,i,j]  (stride L in k)
      a[0] = cosm[(size_t)(b * L_ + kc + klo) * L_ + r0 + l16];
      a[1] = cosm[(size_t)(b * L_ + kc + klo + 1) * L_ + r0 + l16];
    }
    v2f bbv[8];     // batch all fragment loads, then burst the WMMAs
#pragma unroll
    for (int jt = 0; jt < 8; ++jt) {
      int n = n0 + jt * 16 + l16;
      bbv[jt][0] = Bsrc[(size_t)(b * L_ + kc + klo) * H_ + n];
      bbv[jt][1] = Bsrc[(size_t)(b * L_ + kc + klo + 1) * H_ + n];
    }
#pragma unroll
    for (int jt = 0; jt < 8; ++jt) acc[jt] = wmma4(a, bbv[jt], acc[jt]);
  }
#pragma unroll
  for (int jt = 0; jt < 8; ++jt) {
    int n = n0 + jt * 16 + l16;
#pragma unroll
    for (int v = 0; v < 8; ++v) {
      int r = r0 + v + half * 8;
      dst[(size_t)(b * L_ + r) * H_ + n] = acc[jt][v];
    }
  }
}

// ---------------------------------------------------------------- softmax over H (in place)
__global__ void softmax_kernel(float* asum2, float* asum1,
                               const float* __restrict__ m1, const float* __restrict__ m2) {
  int i = blockIdx.x, b = blockIdx.y, side = blockIdx.z;
  float* row = (side == 0 ? asum2 : asum1) + (size_t)(b * L_ + i) * H_;
  float mk = (side == 0 ? m1 : m2)[b * L_ + i];
  int t = threadIdx.x;
  if (mk == 0.f) {  // fully masked row -> uniform
    for (int h = t; h < H_; h += 256) row[h] = 1.0f / (float)H_;
    return;
  }
  __shared__ float red[256];
  float m = NINF;
  for (int h = t; h < H_; h += 256) m = fmaxf(m, row[h]);
  red[t] = m; __syncthreads();
  for (int s = 128; s > 0; s >>= 1) { if (t < s) red[t] = fmaxf(red[t], red[t + s]); __syncthreads(); }
  m = red[0]; __syncthreads();
  float ssum = 0.f;
  for (int h = t; h < H_; h += 256) ssum += expf(row[h] - m);
  red[t] = ssum; __syncthreads();
  for (int s = 128; s > 0; s >>= 1) { if (t < s) red[t] += red[t + s]; __syncthreads(); }
  float inv = 1.0f / red[0];
  for (int h = t; h < H_; h += 256) row[h] = expf(row[h] - m) * inv;
}

// ---------------------------------------------------------------- att elementwise masked max
__global__ void attmax_kernel(const float* __restrict__ cosm,
                              const float* __restrict__ c1, const float* __restrict__ c2,
                              const float* __restrict__ m1, const float* __restrict__ m2,
                              float* amax2, float* amax1) {
  int r = blockIdx.x, b = blockIdx.y, side = blockIdx.z;
  __shared__ float crow[L_], mrow[L_];
  int t = threadIdx.x;
  if (t < L_) {
    if (side == 0) { crow[t] = cosm[(size_t)(b * L_ + r) * L_ + t]; mrow[t] = m2[b * L_ + t]; }
    else           { crow[t] = cosm[(size_t)(b * L_ + t) * L_ + r]; mrow[t] = m1[b * L_ + t]; }
  }
  __syncthreads();
  const float* src = (side == 0) ? c2 : c1;
  float* dst = ((side == 0) ? amax2 : amax1) + (size_t)(b * L_ + r) * H_;
  for (int h = t; h < H_; h += 256) {
    float m = NINF;
    for (int k = 0; k < L_; ++k)
      if (mrow[k] != 0.f) m = fmaxf(m, src[(size_t)(b * L_ + k) * H_ + h] * crow[k]);
    dst[h] = m;
  }
}

// ---------------------------------------------------------------- multi-perspective cosine matches
__global__ void mpm_kernel(const float* __restrict__ c1, const float* __restrict__ c2,
                           const float* __restrict__ asum2, const float* __restrict__ asum1,
                           const float* __restrict__ amax2, const float* __restrict__ amax1,
                           const float* __restrict__ w_ff, const float* __restrict__ w_fb,
                           const float* __restrict__ w_att, const float* __restrict__ w_matt,
                           const float* __restrict__ lens, float* out) {
  int i = blockIdx.x, b = blockIdx.y, z = blockIdx.z;
  int side = z >> 2, tp = z & 3;
  const float* vrow = ((side == 0) ? c1 : c2) + (size_t)(b * L_ + i) * H_;
  const float* xrow;
  if (side == 0) {
    if (tp == 0)      { int lp = (int)(lens[24 + b] + 0.5f); xrow = c2 + (size_t)(b * L_ + lp) * H_; }
    else if (tp == 1) xrow = c2 + (size_t)(b * L_) * H_;
    else if (tp == 2) xrow = asum2 + (size_t)(b * L_ + i) * H_;
    else              xrow = amax2 + (size_t)(b * L_ + i) * H_;
  } else {
    if (tp == 0)      { int lp = (int)(lens[16 + b] + 0.5f); xrow = c1 + (size_t)(b * L_ + lp) * H_; }
    else if (tp == 1) xrow = c1 + (size_t)(b * L_) * H_;
    else if (tp == 2) xrow = asum1 + (size_t)(b * L_ + i) * H_;
    else              xrow = amax1 + (size_t)(b * L_ + i) * H_;
  }
  const float* wm = (tp == 0) ? w_ff : (tp == 1) ? w_fb : (tp == 2) ? w_att : w_matt;
  int colbase = (tp == 0) ? 2 : (tp == 1) ? 19 : (tp == 2) ? 68 : 85;
  __shared__ float sa[H_], sx[H_];
  __shared__ float rd[256], rna[256], rnx[256];
  int t = threadIdx.x;
  float d = 0.f, na = 0.f, nx = 0.f;
  for (int h = t; h < H_; h += 256) {
    float a = vrow[h], x = xrow[h];
    sa[h] = a; sx[h] = x;
    d += a * x; na += a * a; nx += x * x;
  }
  rd[t] = d; rna[t] = na; rnx[t] = nx;
  __syncthreads();
  for (int s = 128; s > 0; s >>= 1) {
    if (t < s) { rd[t] += rd[t + s]; rna[t] += rna[t + s]; rnx[t] += rnx[t + s]; }
    __syncthreads();
  }
  float* o = out + ((size_t)b * 256 + ((side == 0) ? i : 128 + i)) * DCOL;
  if (t == 0)
    o[colbase] = rd[0] / (fmaxf(sqrtf(rna[0]), EPSF) * fmaxf(sqrtf(rnx[0]), EPSF));
  // multi: one 16-lane group per perspective
  int p = t >> 4, l = t & 15;
  float wd = 0.f, wa = 0.f, wx = 0.f;
  for (int h = l; h < H_; h += 16) {
    float w = wm[p * H_ + h]; float w2 = w * w;
    float a = sa[h], x = sx[h];
    wd += w2 * a * x; wa += w2 * a * a; wx += w2 * x * x;
  }
  for (int off = 8; off >= 1; off >>= 1) {
    wd += __shfl_xor(wd, off, 16);
    wa += __shfl_xor(wa, off, 16);
    wx += __shfl_xor(wx, off, 16);
  }
  if (l == 0)
    o[colbase + 1 + p] = wd / (fmaxf(sqrtf(wa), EPSF) * fmaxf(sqrtf(wx), EPSF));
}

// ---------------------------------------------------------------- host driver
extern "C" void kernel_launch(void* const* d_in, const int* in_sizes, int n_in,
                              void* d_out, int out_size, void* d_ws, size_t ws_size,
                              hipStream_t stream) {
  const float* ctx1  = (const float*)d_in[0];
  const float* m1    = (const float*)d_in[1];
  const float* ctx2  = (const float*)d_in[2];
  const float* m2    = (const float*)d_in[3];
  const float* w_ff  = (const float*)d_in[4];
  const float* w_fb  = (const float*)d_in[5];
  const float* w_mp  = (const float*)d_in[6];
  const float* w_att = (const float*)d_in[7];
  const float* w_matt= (const float*)d_in[8];
  float* out = (float*)d_out;

  float* F = (float*)d_ws;
  size_t o = 0;
  float* c1    = F + o; o += (size_t)B_ * L_ * H_;
  float* c2    = F + o; o += (size_t)B_ * L_ * H_;
  float* n1    = F + o; o += (size_t)B_ * L_;
  float* n2    = F + o; o += (size_t)B_ * L_;
  float* lens  = F + o; o += 32;
  float* w2mp  = F + o; o += (size_t)P_ * H_;
  float* wn1   = F + o; o += (size_t)B_ * P_ * L_;
  float* wn2   = F + o; o += (size_t)B_ * P_ * L_;
  float* cosm  = F + o; o += (size_t)B_ * L_ * L_;
  float* asum2 = F + o; o += (size_t)B_ * L_ * H_;
  float* asum1 = F + o; o += (size_t)B_ * L_ * H_;
  float* amax2 = F + o; o += (size_t)B_ * L_ * H_;
  float* amax1 = F + o; o += (size_t)B_ * L_ * H_;
  float* colmax= F + o; o += (size_t)B_ * P_ * 8 * L_;
  float* colsum= F + o; o += (size_t)B_ * P_ * 8 * L_;

  prep_kernel<<<dim3(B_ * L_, 2), 256, 0, stream>>>(ctx1, m1, ctx2, m2, c1, c2, n1, n2);
  lens_kernel<<<1, 32, 0, stream>>>(m1, m2, lens);
  w2_kernel<<<(P_ * H_ + 255) / 256, 256, 0, stream>>>(w_mp, w2mp);
  wn_kernel<<<dim3(B_ * L_, 2), 256, 0, stream>>>(c1, c2, w2mp, wn1, wn2);
  cos_gemm_kernel<<<B_, 256, 0, stream>>>(c1, c2, n1, n2, cosm);
  cosred_kernel<<<dim3(L_, B_, 2), 128, 0, stream>>>(cosm, m1, m2, lens, out);
  mp_gemm_kernel<<<B_ * P_, 256, 0, stream>>>(c1, c2, w2mp, wn1, wn2, m1, m2, lens,
                                              out, colmax, colsum);
  colfinish_kernel<<<(B_ * P_ * L_ + 255) / 256, 256, 0, stream>>>(colmax, colsum, lens, out);
  attsum_gemm_kernel<<<dim3(6, B_, 2), 256, 0, stream>>>(cosm, c1, c2, asum2, asum1);
  softmax_kernel<<<dim3(L_, B_, 2), 256, 0, stream>>>(asum2, asum1, m1, m2);
  attmax_kernel<<<dim3(L_, B_, 2), 256, 0, stream>>>(cosm, c1, c2, m1, m2, amax2, amax1);
  mpm_kernel<<<dim3(L_, B_, 8), 256, 0, stream>>>(c1, c2, asum2, asum1, amax2, amax1,
                                                  w_ff, w_fb, w_att, w_matt, lens, out);
  (void)in_sizes; (void)n_in; (void)out_size; (void)ws_size;
}